// UnlikelihoodLoss_16140487098824
// MI455X (gfx1250) — compile-verified
//
#include <hip/hip_runtime.h>
#include <hip/hip_bf16.h>
#include <cstdint>

// ---------------------------------------------------------------------------
// UnlikelihoodLoss for MI455X (gfx1250, wave32, WGP w/ 320KB LDS).
//
// Plan (memory-roofline driven: 512MiB of logits read exactly once ~22us):
//   k0 init      : first_occ[v]=INT_MAX, counters=0
//   k1 scan      : atomicMin first-occurrence per target token; count norm
//   k2 compact   : deterministic (k-ordered) list of (token, first_occ) pairs
//   k3 row kernel: 1 workgroup per row. Async-copy the 128KB f32 row into
//                  LDS (global_load_async_to_lds_b128, s_wait_asynccnt),
//                  reduce max/sumX/sumExp (WMMA f32 16x16x4 lane-reduction),
//                  then gather neg-token terms from LDS.
//   k4 final     : fixed-order reduction of per-row partials -> scalar.
//
// Workspace layout (needs V*4 + N*16 + 32 bytes ~= 194KB for N=4096,V=32000).
// ---------------------------------------------------------------------------

typedef __attribute__((ext_vector_type(2))) float v2f;
typedef __attribute__((ext_vector_type(8))) float v8f;
typedef int v4i __attribute__((vector_size(16)));
typedef __attribute__((address_space(1))) v4i* as1_v4i;
typedef __attribute__((address_space(3))) v4i* as3_v4i;

// ---- async global->LDS copy (CDNA5 GLOBAL_LOAD_ASYNC_TO_LDS_B128) ----------
__device__ __forceinline__ void async_copy_b128(const float* gsrc, float* ldst) {
#if __has_builtin(__builtin_amdgcn_global_load_async_to_lds_b128)
  __builtin_amdgcn_global_load_async_to_lds_b128((as1_v4i)gsrc, (as3_v4i)ldst,
                                                 0, 0);
#else
  // fallback: synchronous 16B copy (global_load_b128 + ds_store_b128)
  *(float4*)ldst = *(const float4*)gsrc;
#endif
}

__device__ __forceinline__ void wait_async0() {
#if __has_builtin(__builtin_amdgcn_s_wait_asynccnt)
  __builtin_amdgcn_s_wait_asynccnt(0);
#elif __has_builtin(__builtin_amdgcn_global_load_async_to_lds_b128)
  asm volatile("s_wait_asynccnt 0" ::: "memory");
#endif
}

// ---- wave32 pair-reduction via V_WMMA_F32_16X16X4_F32 ----------------------
// A (16x4 f32): lane l<16 holds A[l][0..1] in 2 VGPRs, lane l>=16 holds
// A[l-16][2..3]. We put s in the even-K slot and t in the odd-K slot.
// B (4x16) selector: col0 sums even K (s), col1 sums odd K (t).
// D[m,0] = s_m + s_{m+16}; D[m,1] = t_m + t_{m+16}; finish with 2 shuffles.
__device__ __forceinline__ void wave_reduce_pair(float& s, float& t) {
#if __has_builtin(__builtin_amdgcn_wmma_f32_16x16x4_f32)
  const int lane = threadIdx.x & 31;
  v2f a; a.x = s; a.y = t;
  const int col = lane & 15;
  v2f b; b.x = (col == 0) ? 1.0f : 0.0f;  // B[K even][col]
         b.y = (col == 1) ? 1.0f : 0.0f;  // B[K odd ][col]
  v8f c = {};
  c = __builtin_amdgcn_wmma_f32_16x16x4_f32(false, a, false, b, (short)0, c,
                                            false, false);
  float d = c[0] + c[1] + c[2] + c[3] + c[4] + c[5] + c[6] + c[7];
  // column 0 lives in lanes 0 (M=0..7) and 16 (M=8..15); column 1 in 1 & 17
  s = __shfl(d, 0, 32) + __shfl(d, 16, 32);
  t = __shfl(d, 1, 32) + __shfl(d, 17, 32);
#else
  for (int off = 16; off > 0; off >>= 1) {
    s += __shfl_xor(s, off, 32);
    t += __shfl_xor(t, off, 32);
  }
#endif
}

// ---- k0: init workspace -----------------------------------------------------
__global__ void ul_init_kernel(int* __restrict__ first_occ,
                               int* __restrict__ counters, int V) {
  int v = blockIdx.x * blockDim.x + threadIdx.x;
  if (v < V) first_occ[v] = 0x7fffffff;
  if (v < 4) counters[v] = 0;
}

// ---- k1: first occurrence + norm -------------------------------------------
__global__ void ul_scan_kernel(const int* __restrict__ target,
                               int* __restrict__ first_occ,
                               int* __restrict__ counters, int N) {
  int k = blockIdx.x * blockDim.x + threadIdx.x;
  if (k >= N) return;
  int t = target[k];
  bool valid = (t != -1);
  int tt = valid ? t : 0;                 // jnp.where(valid, target, PAD)
  atomicMin(&first_occ[tt], k);
  if (valid && tt != 0) atomicAdd(&counters[1], 1);  // norm
}

// ---- k2: deterministic compaction (single block, ballot + prefix scan) -----
__global__ void __launch_bounds__(1024)
ul_compact_kernel(const int* __restrict__ target,
                  const int* __restrict__ first_occ, int* __restrict__ list_v,
                  int* __restrict__ list_f, int* __restrict__ counters, int N) {
  __shared__ int wtot[32];
  __shared__ int woff[33];
  __shared__ int base_s;
  const int tid = threadIdx.x, lane = tid & 31, wave = tid >> 5;
  const int nw = blockDim.x >> 5;
  if (tid == 0) base_s = 0;
  __syncthreads();
  for (int chunk = 0; chunk < N; chunk += blockDim.x) {
    int k = chunk + tid;
    int flag = 0, tt = 0;
    if (k < N) {
      int t = target[k];
      tt = (t != -1) ? t : 0;
      flag = (tt != 0 && first_occ[tt] == k) ? 1 : 0;  // PAD never a neg token
    }
    unsigned m32 = (unsigned)__ballot(flag);           // wave32: low 32 bits
    int prefix = __popc(m32 & ((1u << lane) - 1u));
    if (lane == 0) wtot[wave] = __popc(m32);
    __syncthreads();
    if (tid == 0) {
      int r = base_s;
      for (int w = 0; w < nw; ++w) { woff[w] = r; r += wtot[w]; }
      woff[nw] = r;
    }
    __syncthreads();
    if (flag) {
      int pos = woff[wave] + prefix;   // list stays sorted by first-occ index
      list_v[pos] = tt;
      list_f[pos] = k;
    }
    __syncthreads();
    if (tid == 0) base_s = woff[nw];
    __syncthreads();
  }
  if (threadIdx.x == 0) counters[0] = base_s;  // distinct-token count
}

// ---- k3: per-row kernel (row resident in LDS) ------------------------------
__global__ void __launch_bounds__(1024)
ul_row_kernel(const float* __restrict__ logits, const int* __restrict__ target,
              const int* __restrict__ list_v, const int* __restrict__ list_f,
              const int* __restrict__ counters, float* __restrict__ kl_out,
              float* __restrict__ cust_out, int N, int V) {
  extern __shared__ float sm[];      // [0,V): row ; [V, V+80): reduction scratch
  float* red = sm + V;
  const int i = blockIdx.x;
  const int tid = threadIdx.x;
  const int nth = blockDim.x;
  const int lane = tid & 31, wave = tid >> 5, nw = nth >> 5;
  const float* row = logits + (size_t)i * (size_t)V;

  // Phase A0: async DMA the whole row into LDS (bypasses VGPRs).
  const int nvec = V >> 2;
  for (int c = tid; c < nvec; c += nth) async_copy_b128(row + 4 * c, sm + 4 * c);
  for (int v = (nvec << 2) + tid; v < V; v += nth) sm[v] = row[v];  // remainder
  wait_async0();
  __syncthreads();

  // Phase A1: row max and sum(x)
  float mx = -INFINITY, sx = 0.0f;
  for (int v = tid; v < V; v += nth) {
    float x = sm[v];
    mx = fmaxf(mx, x);
    sx += x;
  }
  for (int off = 16; off > 0; off >>= 1) mx = fmaxf(mx, __shfl_xor(mx, off, 32));
  if (lane == 0) red[wave] = mx;
  __syncthreads();
  if (wave == 0) {
    float m = (lane < nw) ? red[lane] : -INFINITY;
    for (int off = 16; off > 0; off >>= 1) m = fmaxf(m, __shfl_xor(m, off, 32));
    if (lane == 0) red[64] = m;
  }
  __syncthreads();
  mx = red[64];

  // Phase A2: sum(exp(x-mx)); reduce (sumExp, sumX) pair via WMMA lane-reduce
  float se = 0.0f;
  for (int v = tid; v < V; v += nth) se += __expf(sm[v] - mx);
  wave_reduce_pair(se, sx);
  if (lane == 0) { red[2 * wave] = se; red[2 * wave + 1] = sx; }
  __syncthreads();
  if (wave == 0) {
    float a = (lane < nw) ? red[2 * lane] : 0.0f;
    float b = (lane < nw) ? red[2 * lane + 1] : 0.0f;
    wave_reduce_pair(a, b);
    if (lane == 0) { red[64] = a; red[65] = b; }
  }
  __syncthreads();
  se = red[64];
  sx = red[65];

  const float logZ = logf(se);
  const float inv_se = 1.0f / se;
  const int t_raw = target[i];
  const bool valid = (t_raw != -1);
  const int tg = valid ? t_raw : 0;

  if (tid == 0) {
    // kl_i = C - SM*sum(logp) - (CONF-SM)*logp[tg]
    const float SM = 0.1f / (float)(V - 1);
    const float C = 0.1f * logf(SM) + 0.9f * logf(0.9f);
    float sumlogp = sx - (float)V * (mx + logZ);
    float logp_t = sm[tg] - mx - logZ;
    float k = C - SM * sumlogp - (0.9f - SM) * logp_t;
    kl_out[i] = valid ? k : 0.0f;
  }

  // Phase B: unlikelihood gathers. list_f ascending -> per-thread early break.
  const int cnt = counters[0];
  float acc = 0.0f;
  for (int e = tid; e < cnt; e += nth) {
    int f = list_f[e];
    if (f >= i) break;                     // sorted: nothing further qualifies
    int vtok = list_v[e];
    if (vtok != tg) {
      float p = __expf(sm[vtok] - mx) * inv_se;   // exp(logp)
      float om = fmaxf(1.0f - p, 1e-5f);          // clip(1-p, 1e-5)
      acc -= logf(om);
    }
  }
  float dummy = 0.0f;
  __syncthreads();                                 // protect scratch reuse
  wave_reduce_pair(acc, dummy);
  if (lane == 0) red[wave] = acc;
  __syncthreads();
  if (wave == 0) {
    float a = (lane < nw) ? red[lane] : 0.0f;
    float b = 0.0f;
    wave_reduce_pair(a, b);
    if (lane == 0) cust_out[i] = valid ? a : 0.0f;
  }
}

// ---- k4: fixed-order final reduction ---------------------------------------
__global__ void __launch_bounds__(256)
ul_final_kernel(const float* __restrict__ kl, const float* __restrict__ cust,
                const int* __restrict__ counters, float* __restrict__ out,
                int N) {
  __shared__ float r1[32], r2[32];
  const int tid = threadIdx.x, lane = tid & 31, wave = tid >> 5;
  const int nw = blockDim.x >> 5;
  float a = 0.0f, b = 0.0f;
  for (int i = tid; i < N; i += blockDim.x) { a += kl[i]; b += cust[i]; }
  for (int off = 16; off > 0; off >>= 1) {
    a += __shfl_xor(a, off, 32);
    b += __shfl_xor(b, off, 32);
  }
  if (lane == 0) { r1[wave] = a; r2[wave] = b; }
  __syncthreads();
  if (wave == 0) {
    a = (lane < nw) ? r1[lane] : 0.0f;
    b = (lane < nw) ? r2[lane] : 0.0f;
    for (int off = 16; off > 0; off >>= 1) {
      a += __shfl_xor(a, off, 32);
      b += __shfl_xor(b, off, 32);
    }
    if (lane == 0) {
      float norm = (float)counters[1];
      out[0] = a / norm + 0.1f * b / norm;
    }
  }
}

// ---------------------------------------------------------------------------
extern "C" void kernel_launch(void* const* d_in, const int* in_sizes, int n_in,
                              void* d_out, int out_size, void* d_ws,
                              size_t ws_size, hipStream_t stream) {
  const float* logits = (const float*)d_in[0];
  const int* target = (const int*)d_in[1];
  const int N = in_sizes[1];
  const int V = in_sizes[0] / N;

  char* ws = (char*)d_ws;
  int* first_occ = (int*)ws;                                   // V ints
  int* list_v = (int*)(ws + (size_t)V * 4);                    // N ints
  int* list_f = (int*)(ws + (size_t)V * 4 + (size_t)N * 4);    // N ints
  float* kl = (float*)(ws + (size_t)V * 4 + (size_t)N * 8);    // N floats
  float* cust = (float*)(ws + (size_t)V * 4 + (size_t)N * 12); // N floats
  int* counters = (int*)(ws + (size_t)V * 4 + (size_t)N * 16); // 4 ints

  ul_init_kernel<<<(V + 255) / 256, 256, 0, stream>>>(first_occ, counters, V);
  ul_scan_kernel<<<(N + 255) / 256, 256, 0, stream>>>(target, first_occ,
                                                      counters, N);
  ul_compact_kernel<<<1, 1024, 0, stream>>>(target, first_occ, list_v, list_f,
                                            counters, N);
  const size_t shmem = ((size_t)V + 80) * sizeof(float);  // 128KB row + scratch
  ul_row_kernel<<<N, 1024, shmem, stream>>>(logits, target, list_v, list_f,
                                            counters, kl, cust, N, V);
  ul_final_kernel<<<1, 256, 0, stream>>>(kl, cust, counters, (float*)d_out, N);
}